// TransformerTreeStack_55336358642356
// MI455X (gfx1250) — compile-verified
//
#include <hip/hip_runtime.h>
#include <math.h>

typedef float v2f __attribute__((ext_vector_type(2)));
typedef float v4f __attribute__((ext_vector_type(4)));
typedef float v8f __attribute__((ext_vector_type(8)));

#define D_DIM 128
#define L_DIM 512

__device__ __forceinline__ v4f nt_load4(const float* p) {
    return __builtin_nontemporal_load((const v4f*)p);
}

// ---------------------------------------------------------------------------
// Kernel 1: QKV projection via fp32 WMMA.
//   out[m, n] = sum_k inputs[m, k] * W[n, k] + bias[n]   (out = in @ W^T + b)
// Grid: (B/16 m-tiles, 3 matrices). Block: 256 threads = 8 waves, one wave
// per 16-wide N tile (D=128 -> 8 tiles). Each wave: 32x v_wmma_f32_16x16x4_f32.
// A fragment (16x4 f32): lane<16 -> M=lane, K={0,1}; lane>=16 -> M=lane-16, K={2,3}
// B fragment (4x16 f32): lane<16 -> N=lane, K={0,1}; lane>=16 -> N=lane-16, K={2,3}
// C/D (16x16 f32): VGPR r -> M = r + 8*(lane>>4), N = lane&15
// ---------------------------------------------------------------------------
__global__ __launch_bounds__(256) void qkv_proj_wmma(
    const float* __restrict__ inputs,
    const float* __restrict__ Wq, const float* __restrict__ bq,
    const float* __restrict__ Wk, const float* __restrict__ bk,
    const float* __restrict__ Wv, const float* __restrict__ bv,
    float* __restrict__ qo, float* __restrict__ ko, float* __restrict__ vo)
{
    const int lane = threadIdx.x & 31;
    const int wv   = threadIdx.x >> 5;       // 0..7 -> N tile
    const int m0   = blockIdx.x * 16;
    const int n0   = wv * 16;
    const int mat  = blockIdx.y;             // 0=q, 1=k, 2=v

    const float* W  = (mat == 0) ? Wq : (mat == 1) ? Wk : Wv;
    const float* bs = (mat == 0) ? bq : (mat == 1) ? bk : bv;
    float*       out = (mat == 0) ? qo : (mat == 1) ? ko : vo;

    const int half = lane >> 4;              // 0 or 1
    const int row  = lane & 15;
    const int kh   = half * 2;               // K offset within 4-chunk

    const float* Ap = inputs + (size_t)(m0 + row) * D_DIM + kh;  // A[m, k]
    const float* Bp = W      + (size_t)(n0 + row) * D_DIM + kh;  // B[k, n] = W[n, k]

    v8f acc = {0.f, 0.f, 0.f, 0.f, 0.f, 0.f, 0.f, 0.f};
#pragma unroll 8
    for (int k0 = 0; k0 < D_DIM; k0 += 4) {
        v2f a = *(const v2f*)(Ap + k0);
        v2f b = *(const v2f*)(Bp + k0);
        acc = __builtin_amdgcn_wmma_f32_16x16x4_f32(
            /*neg_a=*/false, a, /*neg_b=*/false, b,
            /*c_mod=*/(short)0, acc, /*reuse_a=*/false, /*reuse_b=*/false);
    }

    const int   ncol  = n0 + row;
    const float bias  = bs[ncol];
    const int   mbase = m0 + half * 8;
#pragma unroll
    for (int r = 0; r < 8; ++r)
        out[(size_t)(mbase + r) * D_DIM + ncol] = acc[r] + bias;
}

// ---------------------------------------------------------------------------
// Kernel 2: masked stack attention, one block (8 wave32) per batch row.
// Streams only the valid suffix [start, 512) of K and V with non-temporal
// b128 loads (each wave loads one full 512B row, perfectly coalesced).
// ---------------------------------------------------------------------------
__global__ __launch_bounds__(256) void tree_attn(
    const float* __restrict__ prev_keys,
    const float* __restrict__ prev_values,
    const float* __restrict__ qn, const float* __restrict__ kn,
    const float* __restrict__ vn,
    const long long* __restrict__ lengths,
    float* __restrict__ outp)
{
    __shared__ float s_q[D_DIM];
    __shared__ float s_sc[L_DIM + 1];
    __shared__ float s_redm[8];
    __shared__ float s_reds[8];
    __shared__ float s_acc[D_DIM];

    const int b    = blockIdx.x;
    const int tid  = threadIdx.x;
    const int lane = tid & 31;
    const int wv   = tid >> 5;

    const int len   = (int)lengths[b];
    const int start = L_DIM - len;           // first unmasked slot

    if (tid < D_DIM) {
        s_q[tid]   = qn[(size_t)b * D_DIM + tid];
        s_acc[tid] = 0.0f;
    }
    __syncthreads();

    const float inv_sqrt_d = 0.088388347648318447f;   // 1/sqrt(128)
    const v4f qreg = *(const v4f*)(s_q + lane * 4);

    // ---- Phase A: scores (wave-per-row over the valid suffix) ----
    const float* Kb = prev_keys + (size_t)b * L_DIM * D_DIM;
    for (int n = start + wv; n < L_DIM; n += 8) {
        v4f kr = nt_load4(Kb + (size_t)n * D_DIM + lane * 4);
        float p = kr[0]*qreg[0] + kr[1]*qreg[1] + kr[2]*qreg[2] + kr[3]*qreg[3];
#pragma unroll
        for (int off = 16; off > 0; off >>= 1) p += __shfl_xor(p, off, 32);
        if (lane == 0) s_sc[n] = p * inv_sqrt_d;
    }
    if (wv == 0) {  // appended slot (always valid)
        v4f kr = *(const v4f*)(kn + (size_t)b * D_DIM + lane * 4);
        float p = kr[0]*qreg[0] + kr[1]*qreg[1] + kr[2]*qreg[2] + kr[3]*qreg[3];
#pragma unroll
        for (int off = 16; off > 0; off >>= 1) p += __shfl_xor(p, off, 32);
        if (lane == 0) s_sc[L_DIM] = p * inv_sqrt_d;
    }
    __syncthreads();

    // ---- Phase B: block softmax over s_sc[start .. 512] ----
    float mloc = -__builtin_inff();
    for (int n = start + tid; n <= L_DIM; n += 256) mloc = fmaxf(mloc, s_sc[n]);
#pragma unroll
    for (int off = 16; off > 0; off >>= 1) mloc = fmaxf(mloc, __shfl_xor(mloc, off, 32));
    if (lane == 0) s_redm[wv] = mloc;
    __syncthreads();
    float mall = s_redm[0];
#pragma unroll
    for (int i = 1; i < 8; ++i) mall = fmaxf(mall, s_redm[i]);

    float ssum = 0.0f;
    for (int n = start + tid; n <= L_DIM; n += 256) {
        float e = __expf(s_sc[n] - mall);
        s_sc[n] = e;
        ssum += e;
    }
#pragma unroll
    for (int off = 16; off > 0; off >>= 1) ssum += __shfl_xor(ssum, off, 32);
    if (lane == 0) s_reds[wv] = ssum;
    __syncthreads();
    float tot = 0.0f;
#pragma unroll
    for (int i = 0; i < 8; ++i) tot += s_reds[i];
    const float invsum = 1.0f / tot;

    // ---- Phase C: out = sum_n attn[n] * V[n] (wave-per-row, NT streaming) ----
    const float* Vb = prev_values + (size_t)b * L_DIM * D_DIM;
    v4f acc = {0.f, 0.f, 0.f, 0.f};
    for (int n = start + wv; n < L_DIM; n += 8) {
        const float w = s_sc[n];
        v4f vr = nt_load4(Vb + (size_t)n * D_DIM + lane * 4);
        acc[0] += w * vr[0]; acc[1] += w * vr[1];
        acc[2] += w * vr[2]; acc[3] += w * vr[3];
    }
    if (wv == 0) {
        const float w = s_sc[L_DIM];
        v4f vr = *(const v4f*)(vn + (size_t)b * D_DIM + lane * 4);
        acc[0] += w * vr[0]; acc[1] += w * vr[1];
        acc[2] += w * vr[2]; acc[3] += w * vr[3];
    }
    atomicAdd(&s_acc[lane * 4 + 0], acc[0]);   // ds_add_f32, 8-way combine
    atomicAdd(&s_acc[lane * 4 + 1], acc[1]);
    atomicAdd(&s_acc[lane * 4 + 2], acc[2]);
    atomicAdd(&s_acc[lane * 4 + 3], acc[3]);
    __syncthreads();

    if (tid < D_DIM) outp[(size_t)b * D_DIM + tid] = s_acc[tid] * invsum;
}

// ---------------------------------------------------------------------------
extern "C" void kernel_launch(void* const* d_in, const int* in_sizes, int n_in,
                              void* d_out, int out_size, void* d_ws, size_t ws_size,
                              hipStream_t stream) {
    const float*     inputs      = (const float*)d_in[0];
    const float*     prev_keys   = (const float*)d_in[1];
    const float*     prev_values = (const float*)d_in[2];
    const float*     Wq          = (const float*)d_in[3];
    const float*     bq          = (const float*)d_in[4];
    const float*     Wk          = (const float*)d_in[5];
    const float*     bk          = (const float*)d_in[6];
    const float*     Wv          = (const float*)d_in[7];
    const float*     bv          = (const float*)d_in[8];
    const long long* lengths     = (const long long*)d_in[9];
    float*           out         = (float*)d_out;

    const int B = in_sizes[0] / D_DIM;       // 2048

    float* qbuf = (float*)d_ws;              // [B, 128]
    float* kbuf = qbuf + (size_t)B * D_DIM;  // [B, 128]
    float* vbuf = kbuf + (size_t)B * D_DIM;  // [B, 128]

    qkv_proj_wmma<<<dim3(B / 16, 3), 256, 0, stream>>>(
        inputs, Wq, bq, Wk, bk, Wv, bv, qbuf, kbuf, vbuf);

    tree_attn<<<B, 256, 0, stream>>>(
        prev_keys, prev_values, qbuf, kbuf, vbuf, lengths, out);
}